// GATBlock_19576460935445
// MI455X (gfx1250) — compile-verified
//
#include <hip/hip_runtime.h>
#include <hip/hip_bf16.h>

// GAT layer for MI455X (gfx1250, wave32).
// Stage 1: h = x @ W via V_WMMA_F32_16X16X4_F32. 64x16 tile per wave:
//          4 independent accumulator chains (hides WMMA RAW latency),
//          B fragment reused 4x (cuts L2 refetch ~4x).
// Stage 2: per-node attention coefficients (wave-per-node, shfl reductions).
// Stage 3: per-edge leaky-relu logits + segment-max via ordered-uint atomicMax.
// Stage 4: exp + segment-sum via atomicAdd.
// Stage 5: per-edge head-folded aggregation: out[dst] += (1/H)*sum_h alpha_h*h[src,h,:]
//          (4x less atomic traffic than per-head aggregation, no [N,H,C] temp).

typedef __attribute__((ext_vector_type(2))) float v2f;
typedef __attribute__((ext_vector_type(8))) float v8f;

#define DHID 128
#define NHEAD 4
#define CCH 128
#define HC 512  // NHEAD*CCH
#define SLOPE 0.2f

// ---------------- init: d_out = bias broadcast, mkey = 0 (<= key(-inf)), ssum = 0
__global__ void __launch_bounds__(256) gat_init(float* __restrict__ out,
                                                const float* __restrict__ bias,
                                                unsigned* __restrict__ mkey,
                                                float* __restrict__ ssum, int N) {
    int t = blockIdx.x * blockDim.x + threadIdx.x;
    if (t < N * CCH) out[t] = bias[t & (CCH - 1)];
    if (t < N * NHEAD) { mkey[t] = 0u; ssum[t] = 0.0f; }
}

// ---------------- stage 1: h[N,512] = x[N,128] @ W[128,512], f32 WMMA
// Wave computes 64 rows x 16 cols = 4 subtiles of 16x16 with one shared B frag.
// A-frag (16x4 f32): lanes 0-15 row M=lane K={k,k+1}; lanes 16-31 row M=lane-16 K={k+2,k+3}.
// B-frag (4x16 f32): lanes 0-15 col N=lane K={k,k+1}; lanes 16-31 col N=lane-16 K={k+2,k+3}.
// D (16x16 f32, 8 VGPR): vgpr j -> row +j (lanes 0-15) / +j+8 (lanes 16-31), col n0+(lane&15).
__global__ void __launch_bounds__(256) gat_gemm(const float* __restrict__ x,
                                                const float* __restrict__ W,
                                                float* __restrict__ h, int N) {
    int wid  = (blockIdx.x * blockDim.x + threadIdx.x) >> 5;
    int lane = threadIdx.x & 31;
    int mtile = wid >> 5;          // 64 rows per tile; 32 column tiles (512/16)
    int ntile = wid & 31;
    int m0 = mtile * 64, n0 = ntile * 16;
    if (m0 >= N) return;           // wave-uniform

    int l    = lane & 15;
    int kofs = (lane >> 4) << 1;   // 0 for lanes 0-15, 2 for lanes 16-31

    // A row pointers, clamped for dead subtiles (their stores are skipped;
    // N is a multiple of 16 so valid subtiles are always full).
    const float* arow0;
    const float* arow1;
    const float* arow2;
    const float* arow3;
    {
        int r0 = m0 + 0 * 16 + l; if (r0 > N - 1) r0 = N - 1;
        int r1 = m0 + 1 * 16 + l; if (r1 > N - 1) r1 = N - 1;
        int r2 = m0 + 2 * 16 + l; if (r2 > N - 1) r2 = N - 1;
        int r3 = m0 + 3 * 16 + l; if (r3 > N - 1) r3 = N - 1;
        arow0 = x + (size_t)r0 * DHID + kofs;
        arow1 = x + (size_t)r1 * DHID + kofs;
        arow2 = x + (size_t)r2 * DHID + kofs;
        arow3 = x + (size_t)r3 * DHID + kofs;
    }
    const float* bcol = W + (size_t)kofs * HC + n0 + l;

    v8f acc0 = {}, acc1 = {}, acc2 = {}, acc3 = {};
#pragma unroll 4
    for (int k = 0; k < DHID; k += 4) {
        v2f b;
        b.x = bcol[(size_t)k * HC];
        b.y = bcol[(size_t)(k + 1) * HC];
        v2f a0, a1, a2, a3;
        a0.x = arow0[k]; a0.y = arow0[k + 1];
        a1.x = arow1[k]; a1.y = arow1[k + 1];
        a2.x = arow2[k]; a2.y = arow2[k + 1];
        a3.x = arow3[k]; a3.y = arow3[k + 1];
        // 4 independent accumulation chains -> WMMAs can interleave
        acc0 = __builtin_amdgcn_wmma_f32_16x16x4_f32(false, a0, false, b, (short)0, acc0, false, false);
        acc1 = __builtin_amdgcn_wmma_f32_16x16x4_f32(false, a1, false, b, (short)0, acc1, false, false);
        acc2 = __builtin_amdgcn_wmma_f32_16x16x4_f32(false, a2, false, b, (short)0, acc2, false, false);
        acc3 = __builtin_amdgcn_wmma_f32_16x16x4_f32(false, a3, false, b, (short)0, acc3, false, false);
    }

    int rofs = (lane >> 4) << 3;   // 0 or 8
#define STORE_SUB(SUB, ACC)                                                    \
    {                                                                          \
        int rbase = m0 + (SUB) * 16;                                           \
        if (rbase < N) { /* wave-uniform */                                    \
            float* dp = h + (size_t)rbase * HC + n0 + l;                       \
            _Pragma("unroll")                                                  \
            for (int j = 0; j < 8; ++j)                                        \
                dp[(size_t)(j + rofs) * HC] = ACC[j];                          \
        }                                                                      \
    }
    STORE_SUB(0, acc0)
    STORE_SUB(1, acc1)
    STORE_SUB(2, acc2)
    STORE_SUB(3, acc3)
#undef STORE_SUB
}

// ---------------- stage 2: a_src[n,h] = <h[n,h,:], att_src[h,:]>, same for dst.
__global__ void __launch_bounds__(256) gat_att(const float* __restrict__ h,
                                               const float* __restrict__ att_src,
                                               const float* __restrict__ att_dst,
                                               float* __restrict__ a_src,
                                               float* __restrict__ a_dst, int N) {
    int wid  = (blockIdx.x * blockDim.x + threadIdx.x) >> 5;
    int lane = threadIdx.x & 31;
    if (wid >= N) return;
    const float* hp = h + (size_t)wid * HC;

    float ss[NHEAD], sd[NHEAD];
#pragma unroll
    for (int hh = 0; hh < NHEAD; ++hh) {
        float s1 = 0.f, s2 = 0.f;
#pragma unroll
        for (int i = 0; i < 4; ++i) {
            int cc = i * 32 + lane;            // channel within head
            float v = hp[hh * CCH + cc];
            s1 += v * att_src[hh * CCH + cc];
            s2 += v * att_dst[hh * CCH + cc];
        }
        ss[hh] = s1; sd[hh] = s2;
    }
#pragma unroll
    for (int hh = 0; hh < NHEAD; ++hh) {
        float s1 = ss[hh], s2 = sd[hh];
#pragma unroll
        for (int off = 16; off > 0; off >>= 1) {
            s1 += __shfl_xor(s1, off, 32);
            s2 += __shfl_xor(s2, off, 32);
        }
        if (lane == 0) {
            a_src[wid * NHEAD + hh] = s1;
            a_dst[wid * NHEAD + hh] = s2;
        }
    }
}

// ordered-uint key for float atomic max: monotonic unsigned mapping
__device__ __forceinline__ unsigned fkey(float f) {
    unsigned b = __float_as_uint(f);
    return b ^ ((b & 0x80000000u) ? 0xFFFFFFFFu : 0x80000000u);
}
__device__ __forceinline__ float funkey(unsigned k) {
    return __uint_as_float((k & 0x80000000u) ? (k ^ 0x80000000u) : ~k);
}

// ---------------- stage 3: logits + leaky relu + segment max (per edge,head)
__global__ void __launch_bounds__(256) gat_logits(const int* __restrict__ ei,
                                                  const float* __restrict__ a_src,
                                                  const float* __restrict__ a_dst,
                                                  float* __restrict__ ep,
                                                  unsigned* __restrict__ mkey,
                                                  int E, int N) {
    int t = blockIdx.x * blockDim.x + threadIdx.x;
    int e = t >> 2, hh = t & 3;
    if (e >= E + N) return;
    int s, d;
    if (e < E) { s = ei[e]; d = ei[E + e]; } else { s = d = e - E; }  // self loops
    float v = a_src[s * NHEAD + hh] + a_dst[d * NHEAD + hh];
    v = (v >= 0.f) ? v : SLOPE * v;
    ep[t] = v;
    atomicMax(mkey + d * NHEAD + hh, fkey(v));
}

// ---------------- stage 4: p = exp(e - m[dst]); segment sum
__global__ void __launch_bounds__(256) gat_expsum(const int* __restrict__ ei,
                                                  float* __restrict__ ep,
                                                  const unsigned* __restrict__ mkey,
                                                  float* __restrict__ ssum,
                                                  int E, int N) {
    int t = blockIdx.x * blockDim.x + threadIdx.x;
    int e = t >> 2, hh = t & 3;
    if (e >= E + N) return;
    int d = (e < E) ? ei[E + e] : (e - E);
    float m = funkey(mkey[d * NHEAD + hh]);
    float p = __expf(ep[t] - m);
    ep[t] = p;                                  // overwrite logits with numerators
    atomicAdd(ssum + d * NHEAD + hh, p);
}

// ---------------- stage 5: out[dst,c] += (1/H) * sum_h alpha_h * h[src,h,c]
// One wave per edge: lane owns 4 consecutive channels per head (float4 gather,
// 512B contiguous per wave per head), folds the head-mean before the atomic.
__global__ void __launch_bounds__(256) gat_aggregate(const int* __restrict__ ei,
                                                     const float* __restrict__ h,
                                                     const float* __restrict__ ep,
                                                     const float* __restrict__ ssum,
                                                     float* __restrict__ out,
                                                     int E, int N) {
    int wid  = (blockIdx.x * blockDim.x + threadIdx.x) >> 5;
    int lane = threadIdx.x & 31;
    if (wid >= E + N) return;
    int s, d;
    if (wid < E) { s = ei[wid]; d = ei[E + wid]; } else { s = d = wid - E; }

    float alpha[NHEAD];
#pragma unroll
    for (int hh = 0; hh < NHEAD; ++hh)
        alpha[hh] = 0.25f * ep[(size_t)wid * NHEAD + hh] / ssum[d * NHEAD + hh];

    const float4* hs = (const float4*)(h + (size_t)s * HC);
    float4 acc = {0.f, 0.f, 0.f, 0.f};
#pragma unroll
    for (int hh = 0; hh < NHEAD; ++hh) {
        float4 v = hs[hh * 32 + lane];
        acc.x += alpha[hh] * v.x;
        acc.y += alpha[hh] * v.y;
        acc.z += alpha[hh] * v.z;
        acc.w += alpha[hh] * v.w;
    }
    float* op = out + (size_t)d * CCH + lane * 4;
    atomicAdd(op + 0, acc.x);
    atomicAdd(op + 1, acc.y);
    atomicAdd(op + 2, acc.z);
    atomicAdd(op + 3, acc.w);
}

extern "C" void kernel_launch(void* const* d_in, const int* in_sizes, int n_in,
                              void* d_out, int out_size, void* d_ws, size_t ws_size,
                              hipStream_t stream) {
    const float* x       = (const float*)d_in[0];
    const int*   ei      = (const int*)d_in[1];
    const float* W       = (const float*)d_in[2];
    const float* att_src = (const float*)d_in[3];
    const float* att_dst = (const float*)d_in[4];
    const float* bias    = (const float*)d_in[5];
    float* out = (float*)d_out;

    int N = in_sizes[0] / DHID;   // 50000
    int E = in_sizes[1] / 2;      // 800000
    int Etot = E + N;

    // workspace layout (floats): h | a_src | a_dst | mkey | ssum | ep
    float*    h     = (float*)d_ws;
    float*    a_src = h + (size_t)N * HC;
    float*    a_dst = a_src + (size_t)N * NHEAD;
    unsigned* mkey  = (unsigned*)(a_dst + (size_t)N * NHEAD);
    float*    ssum  = (float*)(mkey + (size_t)N * NHEAD);
    float*    ep    = ssum + (size_t)N * NHEAD;

    gat_init<<<(N * CCH + 255) / 256, 256, 0, stream>>>(out, bias, mkey, ssum, N);

    int mtiles64 = (N + 63) / 64;
    int tiles    = mtiles64 * (HC / 16);      // waves needed (64x16 per wave)
    gat_gemm<<<(tiles + 7) / 8, 256, 0, stream>>>(x, W, h, N);

    gat_att<<<(N + 7) / 8, 256, 0, stream>>>(h, att_src, att_dst, a_src, a_dst, N);

    int eh = Etot * NHEAD;
    gat_logits<<<(eh + 255) / 256, 256, 0, stream>>>(ei, a_src, a_dst, ep, mkey, E, N);
    gat_expsum<<<(eh + 255) / 256, 256, 0, stream>>>(ei, ep, mkey, ssum, E, N);

    gat_aggregate<<<(Etot + 7) / 8, 256, 0, stream>>>(ei, h, ep, ssum, out, E, N);
}